// Encoder_29454885716042
// MI455X (gfx1250) — compile-verified
//
#include <hip/hip_runtime.h>
#include <stdint.h>

// Problem constants (match reference)
#define BB 4
#define LL 4096
#define DD 512
#define TMAX_INV (1.0f / 200.0f)
#define SIGMA2 100.0f
#define INV_LAMBD2 40000.0f   // 1 / 0.005^2
#define SQRT_D 22.62741699796952f  // sqrt(512)

typedef uint32_t u32x4 __attribute__((ext_vector_type(4)));
typedef int32_t  i32x4 __attribute__((ext_vector_type(4)));
typedef int32_t  i32x8 __attribute__((ext_vector_type(8)));
typedef float    f32x4 __attribute__((ext_vector_type(4)));   // native vec for b128 + NT stores

// One workgroup (256 threads = 8 wave32) per scores row (b,i).
// Single-pass fused masked-softmax: row max is analytically exactly 100.0
// (diagonal element), masked entries underflow to 0 after exp(-1100).
__global__ __launch_bounds__(256) void scores_kernel(const float* __restrict__ event_time,
                                                     float* __restrict__ out)
{
    __shared__ alignas(16) float s_t[LL];   // 16 KB: event_time[b][0..L-1]
    __shared__ float s_red[8];

    const int row = blockIdx.x;        // 0 .. B*L-1
    const int b   = row >> 12;         // row / L
    const int i   = row & (LL - 1);    // row % L
    const int tid = threadIdx.x;

    // ---- CDNA5 Tensor Data Mover: DMA the 16KB time row into LDS ----
    // Wave-uniform branch: only wave 0 issues (TDM ignores EXEC, so gate by branch).
    if (tid < 32) {
        uint64_t gaddr    = (uint64_t)(uintptr_t)(event_time + (size_t)b * LL);
        uint32_t lds_base = (uint32_t)(uintptr_t)(&s_t[0]);  // low 32 bits of generic ptr = LDS offset

        // D# group 0 (ISA 8.3): count=1 | lds_addr | global_addr[56:0] | type=2
        u32x4 g0;
        g0[0] = 1u;
        g0[1] = lds_base;
        g0[2] = (uint32_t)(gaddr & 0xFFFFFFFFu);
        g0[3] = (uint32_t)((gaddr >> 32) & 0x1FFFFFFu) | (2u << 30);

        // D# group 1 (ISA 8.4): data_size=4B; tensor 4096x1; tile 4096x1; stride 4096
        i32x8 g1;
        g1[0] = (int)(2u << 16);                 // workgroup_mask=0, data_size=2 (4B)
        g1[1] = (int)((LL & 0xFFFF) << 16);      // tensor_dim0[15:0] in bits[31:16]
        g1[2] = (int)(((LL >> 16) & 0xFFFF) | (1u << 16)); // tensor_dim0[31:16] | tensor_dim1=1
        g1[3] = (int)((LL & 0xFFFF) << 16);      // tile_dim0 = 4096 in bits[31:16]
        g1[4] = 1;                               // tile_dim1 = 1, tile_dim2 = 0
        g1[5] = LL;                              // tensor_dim0_stride[31:0]
        g1[6] = 0;
        g1[7] = 0;

        i32x4 z4 = {0, 0, 0, 0};                 // groups 2/3 unused (2D tensor)
        i32x8 z8 = {0, 0, 0, 0, 0, 0, 0, 0};     // 5th descriptor group (clang-23 form)
        __builtin_amdgcn_tensor_load_to_lds(g0, g1, z4, z4, z8, 0);
        __builtin_amdgcn_s_wait_tensorcnt(0);
    }
    __syncthreads();

    const float ti = s_t[i] * TMAX_INV;
    const f32x4* __restrict__ s_t4 = (const f32x4*)s_t;

    // Each thread owns 4 groups of 4 consecutive elements (ds_load_b128 in,
    // global_store_b128 out; each wave store covers 512 contiguous bytes).
    f32x4 pv[4];
    float lsum = 0.0f;
#pragma unroll
    for (int k = 0; k < 4; ++k) {
        const int q  = (k << 8) + tid;     // float4 index within row
        const int j0 = q << 2;             // element index
        const f32x4 t4 = s_t4[q];
        f32x4 r;
#pragma unroll
        for (int e = 0; e < 4; ++e) {
            const float dl = ti - t4[e] * TMAX_INV;
            const float d2 = dl * dl;
            const float s  = SIGMA2 * __expf(-(d2 * d2) * INV_LAMBD2);
            const float v  = __expf(s - SIGMA2);   // row max == 100.0 exactly
            r[e] = (j0 + e <= i) ? v : 0.0f;       // masked: exp(-1100) underflows to 0
            lsum += r[e];
        }
        pv[k] = r;
    }

    // wave32 reduction, then cross-wave via LDS
#pragma unroll
    for (int off = 16; off > 0; off >>= 1)
        lsum += __shfl_down(lsum, off, 32);
    if ((tid & 31) == 0) s_red[tid >> 5] = lsum;
    __syncthreads();
    if (tid == 0) {
        float d = 0.0f;
#pragma unroll
        for (int w = 0; w < 8; ++w) d += s_red[w];
        s_red[0] = d;
    }
    __syncthreads();
    const float inv = 1.0f / s_red[0];

    f32x4* __restrict__ orow4 = (f32x4*)(out + (size_t)row * LL);
#pragma unroll
    for (int k = 0; k < 4; ++k) {
        const int q = (k << 8) + tid;
        __builtin_nontemporal_store(pv[k] * inv, &orow4[q]);   // write-once stream: TH_NT
    }
}

// embedding: [B, L, 2D]; d<512 sinusoidal temporal enc, d>=512 gathered type emb * sqrt(D)
// One f32x4 (4 consecutive channels) per thread -> b128 stores.
__global__ __launch_bounds__(256) void embed_kernel(const int* __restrict__ event_type,
                                                    const float* __restrict__ event_time,
                                                    const float* __restrict__ emb_table,
                                                    float* __restrict__ out)
{
    const int g  = blockIdx.x * 256 + threadIdx.x;   // float4 index, 0 .. B*L*256-1
    const int bl = g >> 8;                           // 1024 floats = 256 float4 per (b,l)
    const int d0 = (g & 255) << 2;                   // channel of first element

    f32x4 v;
    if (d0 < DD) {
        const float t = event_time[bl];
        // channels d0..d0+3: pairs (sin,cos) share a frequency; two pows per float4
        const float e0 = (float)d0       * (1.0f / (float)DD);  // 2*(d0/2)/D, d0 even
        const float e1 = (float)(d0 + 2) * (1.0f / (float)DD);
        const float a0 = t / __powf(10000.0f, e0);
        const float a1 = t / __powf(10000.0f, e1);
        v = (f32x4){__sinf(a0), __cosf(a0), __sinf(a1), __cosf(a1)};
    } else {
        const int ty = event_type[bl];
        const f32x4 w = *(const f32x4*)(emb_table + ty * DD + (d0 - DD));
        v = w * SQRT_D;
    }
    __builtin_nontemporal_store(v, (f32x4*)(out) + g);
}

extern "C" void kernel_launch(void* const* d_in, const int* in_sizes, int n_in,
                              void* d_out, int out_size, void* d_ws, size_t ws_size,
                              hipStream_t stream)
{
    const int*   event_type = (const int*)d_in[0];
    const float* event_time = (const float*)d_in[1];
    const float* emb_table  = (const float*)d_in[2];

    float* out    = (float*)d_out;
    float* scores = out;                               // B*L*L floats
    float* embed  = out + (size_t)BB * LL * LL;        // then B*L*2D floats

    scores_kernel<<<BB * LL, 256, 0, stream>>>(event_time, scores);
    embed_kernel<<<(BB * LL * 2 * DD) / 1024, 256, 0, stream>>>(event_type, event_time,
                                                                emb_table, embed);
}